// GraphAttention_10144712753219
// MI455X (gfx1250) — compile-verified
//
#include <hip/hip_runtime.h>

// Problem constants (from the reference)
#define BB   8
#define CC   64
#define NN   4096
#define KK   32
#define OUTC 64

typedef __attribute__((ext_vector_type(2))) float v2f;
typedef __attribute__((ext_vector_type(4))) float v4f;
typedef __attribute__((ext_vector_type(8))) float v8f;

// LDS layout: padded strides so wave32 v2f loads hit all 64 banks.
// stride 132: (132*row) mod 64 == 4*row  -> lanes 0..15 cover banks {4r+c,4r+c+1},
// lanes 16..31 cover the complementary set -> conflict-free.
#define WSTRIDE 132
#define DSTRIDE 132
#define W_ELEMS (OUTC * WSTRIDE)          // 8448 floats
#define D_ELEMS (KK * DSTRIDE)            // 4224 floats
#define WAVES   8
#define SMEM_FLOATS (W_ELEMS + WAVES * D_ELEMS)
#define SMEM_BYTES  (SMEM_FLOATS * 4)     // 168960 B < 320 KB WGP LDS

__global__ __launch_bounds__(256, 1)
void graph_attn_wmma(const int*   __restrict__ gidx,     // (B,N,K)
                     const float* __restrict__ x,        // (B,C,N) == [B][N][C] flat
                     const float* __restrict__ feature,  // (B,OUT,N,K)
                     const float* __restrict__ Wm,       // (OUT, 2C)
                     const float* __restrict__ gamma,
                     const float* __restrict__ beta,
                     const float* __restrict__ bn_mean,
                     const float* __restrict__ bn_var,
                     float*       __restrict__ out)      // (B,OUT,N)
{
    extern __shared__ float smem[];
    float* lw = smem;                       // W in LDS, [OUTC][WSTRIDE]
    const int tid  = threadIdx.x;
    const int lane = tid & 31;
    const int wv   = tid >> 5;              // wave id in block (0..7)
    const int lr   = lane & 15;
    const int hi   = lane >> 4;             // lane half (WMMA layout)
    float* dl = smem + W_ELEMS + wv * D_ELEMS;   // this wave's delta_f [KK][DSTRIDE]

    // ---- cooperative load of W (64 x 128) into LDS (coalesced) ----
    for (int i = tid; i < OUTC * 2 * CC; i += 256) {
        int o = i >> 7;
        int c = i & 127;
        lw[o * WSTRIDE + c] = Wm[i];
    }

    // ---- this wave's point ----
    const int pid = blockIdx.x * WAVES + wv;     // 0 .. 32767
    const int b   = pid >> 12;
    const int n   = pid & (NN - 1);

    // ---- gather neighbors; build delta_f = [centre-nb | nb] (32 x 128) ----
    const float* xb = x + (size_t)b * CC * NN;   // [N][C]
    const int myidx = gidx[(size_t)pid * KK + lane];     // neighbor idx for k=lane
    v2f cen = *(const v2f*)(xb + (size_t)n * CC + 2 * lane);
    for (int r = 0; r < KK; ++r) {
        int nr = __shfl(myidx, r, 32);
        v2f nb = *(const v2f*)(xb + (size_t)nr * CC + 2 * lane);  // coalesced 256B row
        *(v2f*)(dl + r * DSTRIDE + 2 * lane)      = cen - nb;     // cols 0..63
        *(v2f*)(dl + r * DSTRIDE + CC + 2 * lane) = nb;           // cols 64..127
    }
    __syncthreads();   // publish W (delta is wave-local, DS ops are in-order)

    // ---- e[32x64] = delta[32x128] * W^T via V_WMMA_F32_16X16X4_F32 ----
    // A 16x4 f32 layout: lane(lr,hi) holds A[lr][2*hi], A[lr][2*hi+1]
    // B 4x16 f32 layout: lane(lr,hi) holds B[2*hi][lr], B[2*hi+1][lr] = W[o][c],W[o][c+1]
    v8f acc[2][4] = {};
    const int colA = 2 * hi;
    for (int kk = 0; kk < 2 * CC; kk += 4) {
        v2f a0 = *(const v2f*)(dl + lr * DSTRIDE + kk + colA);          // rows 0..15
        v2f a1 = *(const v2f*)(dl + (16 + lr) * DSTRIDE + kk + colA);   // rows 16..31
#pragma unroll
        for (int nt = 0; nt < 4; ++nt) {
            v2f bf = *(const v2f*)(lw + (nt * 16 + lr) * WSTRIDE + kk + colA);
            acc[0][nt] = __builtin_amdgcn_wmma_f32_16x16x4_f32(
                false, a0, false, bf, (short)0, acc[0][nt], false, false);
            acc[1][nt] = __builtin_amdgcn_wmma_f32_16x16x4_f32(
                false, a1, false, bf, (short)0, acc[1][nt], false, false);
        }
    }

    // ---- BN + LeakyReLU + softmax over K + weighted sum with feature ----
    // C/D layout: VGPR j -> (M = j + 8*hi, N = lr) within a 16x16 tile.
    // Lane l and l^16 share column o and split K rows -> one shfl_xor(16) reduces K.
    const float* fbase = feature + ((size_t)b * OUTC * NN + n) * KK;   // + o*NN*KK
    const size_t obase = (size_t)b * OUTC * NN + (size_t)n;            // + o*NN
#pragma unroll
    for (int nt = 0; nt < 4; ++nt) {
        const int o = nt * 16 + lr;
        const float iv = gamma[o] * rsqrtf(bn_var[o] + 1e-5f);
        const float bs = beta[o] - bn_mean[o] * iv;

        float t[2][8];
        float m = -3.402823466e+38f;
#pragma unroll
        for (int h = 0; h < 2; ++h)
#pragma unroll
            for (int j = 0; j < 8; ++j) {
                float u = acc[h][nt][j] * iv + bs;
                u = (u >= 0.0f) ? u : 0.2f * u;      // LeakyReLU(0.2)
                t[h][j] = u;
                m = fmaxf(m, u);
            }
        m = fmaxf(m, __shfl_xor(m, 16, 32));

        float s = 0.0f;
#pragma unroll
        for (int h = 0; h < 2; ++h)
#pragma unroll
            for (int j = 0; j < 8; ++j) {
                float p = __expf(t[h][j] - m);
                t[h][j] = p;
                s += p;
            }
        s += __shfl_xor(s, 16, 32);
        const float rs = 1.0f / s;

        // weighted sum: this lane's k values are {h*16 + 8*hi + j}, contiguous in j
        const float* fo = fbase + (size_t)o * NN * KK;
        float part = 0.0f;
#pragma unroll
        for (int h = 0; h < 2; ++h) {
            const float* fp = fo + h * 16 + 8 * hi;  // 16B-aligned, 32B contiguous
            v4f fa = *(const v4f*)fp;
            v4f fb = *(const v4f*)(fp + 4);
            part += t[h][0] * fa.x + t[h][1] * fa.y + t[h][2] * fa.z + t[h][3] * fa.w
                  + t[h][4] * fb.x + t[h][5] * fb.y + t[h][6] * fb.z + t[h][7] * fb.w;
        }
        part *= rs;
        part += __shfl_xor(part, 16, 32);
        if (hi == 0) out[obase + (size_t)o * NN] = part;
    }
}

extern "C" void kernel_launch(void* const* d_in, const int* in_sizes, int n_in,
                              void* d_out, int out_size, void* d_ws, size_t ws_size,
                              hipStream_t stream) {
    (void)in_sizes; (void)n_in; (void)out_size; (void)d_ws; (void)ws_size;
    const int*   gidx  = (const int*)  d_in[0];
    const float* x     = (const float*)d_in[1];
    const float* feat  = (const float*)d_in[2];
    const float* Wm    = (const float*)d_in[3];
    const float* gam   = (const float*)d_in[4];
    const float* bet   = (const float*)d_in[5];
    const float* bmean = (const float*)d_in[6];
    const float* bvar  = (const float*)d_in[7];
    float* out = (float*)d_out;

    // Allow >64KB dynamic LDS (gfx1250 WGP has 320KB). Safe to call every time.
    (void)hipFuncSetAttribute((const void*)graph_attn_wmma,
                              hipFuncAttributeMaxDynamicSharedMemorySize, SMEM_BYTES);

    dim3 grid(BB * NN / WAVES);   // 4096 blocks x 8 waves = 32768 points
    graph_attn_wmma<<<grid, 256, SMEM_BYTES, stream>>>(
        gidx, x, feat, Wm, gam, bet, bmean, bvar, out);
}